// TypeConcatSheafLearner_31842887533251
// MI455X (gfx1250) — compile-verified
//
#include <hip/hip_runtime.h>
#include <cmath>

typedef float v2f __attribute__((ext_vector_type(2)));
typedef float v8f __attribute__((ext_vector_type(8)));

#define NUM_K_STEPS 37   // 148 / 4 exactly
#define C_DIM 64
#define IN_DIM 148
#define OUT_DIM 9

// One wave (32 lanes) processes a tile of 16 edges with 37 chained
// V_WMMA_F32_16X16X4_F32 ops: [16 edges x 148] @ [148 x 16(9 used)].
// One-hot (src_type, dst_type, edge_type) features are folded into the
// last 5 K-steps, so the whole 148-K contraction is one uniform WMMA loop.
__global__ __launch_bounds__(256)
void sheaf_wmma_kernel(const float* __restrict__ x,
                       const int*   __restrict__ edge_index,   // [2, E] int32
                       const int*   __restrict__ edge_types,   // [E]
                       const int*   __restrict__ node_types,   // [N]
                       const float* __restrict__ W,            // [9, 148]
                       float* __restrict__ out,                // [E, 9]
                       int E)
{
    // B-fragment table: per step s (K=4s..4s+3) and lane l, the two W^T
    // values that lane needs. Layout matches WMMA B layout for 4x16 f32:
    // lanes 0-15 hold rows K=4s,4s+1 for col N=l; lanes 16-31 hold K=4s+2,4s+3.
    __shared__ v2f Btab[NUM_K_STEPS * 32];

    const int tid = threadIdx.x;
    for (int t = tid; t < NUM_K_STEPS * 32; t += blockDim.x) {
        const int n    = t & 15;          // output column (N)
        const int half = (t >> 4) & 1;    // lane group
        const int s    = t >> 5;          // K step
        const int k    = 4 * s + 2 * half;
        v2f b;
        b[0] = (n < OUT_DIM)                      ? W[n * IN_DIM + k]     : 0.0f;
        b[1] = (n < OUT_DIM && (k + 1) < IN_DIM)  ? W[n * IN_DIM + k + 1] : 0.0f;
        Btab[t] = b;
    }
    __syncthreads();

    const int wave = tid >> 5;
    const int lane = tid & 31;
    const int half = lane >> 4;   // 0: K%4 in {0,1}, 1: K%4 in {2,3}
    const int m    = lane & 15;   // edge row within tile / output column

    const int numTiles = (E + 15) >> 4;
    const int tile = blockIdx.x * 8 + wave;   // wave-uniform
    if (tile >= numTiles) return;             // uniform exit, EXEC stays full

    const long long tileBase = (long long)tile * 16;
    long long e = tileBase + m;
    if (e >= E) e = E - 1;                    // clamp (no divergence)

    const long long src = (long long)edge_index[e];
    const long long dst = (long long)edge_index[(long long)E + e];
    const int et = edge_types[e];
    const int st = node_types[src];
    const int dt = node_types[dst];

    const float* xs = x + src * C_DIM;
    const float* xd = x + dst * C_DIM;

    v8f c = {0.f, 0.f, 0.f, 0.f, 0.f, 0.f, 0.f, 0.f};

#pragma unroll
    for (int s = 0; s < NUM_K_STEPS; ++s) {
        v2f a;
        if (s < 16) {                         // x_src features, K = 0..63
            const int k = 4 * s + 2 * half;
            a = *(const v2f*)(xs + k);        // 8B-aligned global_load_b64
        } else if (s < 32) {                  // x_dst features, K = 64..127
            const int k = 4 * (s - 16) + 2 * half;
            a = *(const v2f*)(xd + k);
        } else {                              // one-hot features, K = 128..147
            const int f = 4 * s - 128 + 2 * half;  // 0..19
            const int g = f + 1;
            a[0] = (f < 4) ? (st == f     ? 1.f : 0.f)
                 : (f < 8) ? (dt == f - 4 ? 1.f : 0.f)
                 :           (et == f - 8 ? 1.f : 0.f);
            a[1] = (g < 4) ? (st == g     ? 1.f : 0.f)
                 : (g < 8) ? (dt == g - 4 ? 1.f : 0.f)
                 :           (et == g - 8 ? 1.f : 0.f);
        }
        const v2f b = Btab[s * 32 + lane];    // ds_load_b64
        c = __builtin_amdgcn_wmma_f32_16x16x4_f32(
                /*neg_a=*/false, a, /*neg_b=*/false, b,
                /*c_mod=*/(short)0, c, /*reuse_a=*/false, /*reuse_b=*/false);
    }

    // C/D layout: VGPR j holds row M = j + 8*half, column N = lane%16.
    if (m < OUT_DIM) {
#pragma unroll
        for (int j = 0; j < 8; ++j) {
            const long long erow = tileBase + j + 8 * half;
            if (erow < E) out[erow * OUT_DIM + m] = tanhf(c[j]);
        }
    }
}

extern "C" void kernel_launch(void* const* d_in, const int* in_sizes, int n_in,
                              void* d_out, int out_size, void* d_ws, size_t ws_size,
                              hipStream_t stream) {
    const float* x          = (const float*)d_in[0];
    const int*   edge_index = (const int*)  d_in[1];  // JAX x64 off -> int32
    const int*   edge_types = (const int*)  d_in[2];
    const int*   node_types = (const int*)  d_in[3];
    const float* W          = (const float*)d_in[4];
    float* out = (float*)d_out;

    const int E = in_sizes[2];                 // edge_types element count
    const int numTiles = (E + 15) / 16;
    const int blocks = (numTiles + 7) / 8;     // 8 waves (tiles) per block
    if (blocks > 0) {
        sheaf_wmma_kernel<<<blocks, 256, 0, stream>>>(
            x, edge_index, edge_types, node_types, W, out, E);
    }
}